// similarity_model_51067161149970
// MI455X (gfx1250) — compile-verified
//
#include <hip/hip_runtime.h>

// similarity_model: q = emb[wordid]; sim = emb @ q; top-(k+1) values+indices.
// Memory-bound GEMV (240 MB read ~ 10.3 us @ 23.3 TB/s). fp32 WMMA
// (V_WMMA_F32_16X16X4_F32) does the MACs in the matrix pipe without any
// precision downgrade vs the fp32 reference.

#define VOCAB 200000
#define DIM   300
#define TOPK  11          // reference returns topk+1 = 11 entries
#define S1_BLOCKS 32
#define S1_THREADS 256

typedef __attribute__((ext_vector_type(2))) float v2f;
typedef __attribute__((ext_vector_type(8))) float v8f;

// ---------------------------------------------------------------------------
// Kernel 1: sim[v] = dot(emb[v], emb[wordid]) via V_WMMA_F32_16X16X4_F32.
// One wave handles 16 vocab rows. A-matrix (16x4 f32): lane m (0..15) holds
// row m, K=0..1; lane 16+m holds row m, K=2..3 (CDNA5 ISA layout). B is the
// query replicated into all 16 columns, so every column of C is the same
// 16-vector of dot products; lanes with N==0 (lane 0 -> M=0..7, lane 16 ->
// M=8..15) write the results.
// ---------------------------------------------------------------------------
__global__ __launch_bounds__(128) void gemv_wmma(const float* __restrict__ emb,
                                                 const int*   __restrict__ wid,
                                                 float*       __restrict__ sim) {
  const int lane    = threadIdx.x & 31;
  const int wave    = blockIdx.x * (blockDim.x >> 5) + (threadIdx.x >> 5);
  const int rowBase = wave * 16;               // 12500 waves * 16 = 200000 exactly
  const int m       = lane & 15;
  const int koff    = (lane >> 4) << 1;        // 0 for lanes 0-15, 2 for 16-31

  const float* q      = emb + (size_t)wid[0] * DIM;
  const float* rowPtr = emb + (size_t)(rowBase + m) * DIM + koff;
  const float* qPtr   = q + koff;

  v8f c = {};
  #pragma unroll 5
  for (int k = 0; k < DIM; k += 4) {           // 75 WMMA steps, K=4 each
    v2f a = *(const v2f*)(rowPtr + k);         // global_load_b64 (8B aligned)
    v2f b = *(const v2f*)(qPtr + k);           // broadcast, stays in WGP$
    c = __builtin_amdgcn_wmma_f32_16x16x4_f32(
        /*neg_a=*/false, a, /*neg_b=*/false, b,
        /*c_mod=*/(short)0, c, /*reuse_a=*/false, /*reuse_b=*/false);
  }

  if (m == 0) {                                // lane 0 and lane 16: column N=0
    const int base = rowBase + (lane >> 4) * 8;
    #pragma unroll
    for (int r = 0; r < 8; ++r) sim[base + r] = c[r];
  }
}

// ---------------------------------------------------------------------------
// Top-k helpers: sorted-descending 11-lists, insertion + pairwise merge.
// ---------------------------------------------------------------------------
__device__ inline void insert_desc(float (&v)[TOPK], int (&id)[TOPK],
                                   float nv, int nid) {
  if (!(nv > v[TOPK - 1])) return;   // strict >: earlier (smaller) index wins ties
  int j = TOPK - 1;
  while (j > 0 && v[j - 1] < nv) { v[j] = v[j - 1]; id[j] = id[j - 1]; --j; }
  v[j] = nv; id[j] = nid;
}

__device__ inline void merge_desc(float (&a)[TOPK], int (&ai)[TOPK],
                                  const float (&b)[TOPK], const int (&bi)[TOPK]) {
  float o[TOPK]; int oi[TOPK];
  int p = 0, q = 0;
  for (int t = 0; t < TOPK; ++t) {
    bool takeA;
    if (p >= TOPK)       takeA = false;
    else if (q >= TOPK)  takeA = true;
    else takeA = (a[p] > b[q]) || (a[p] == b[q] && ai[p] <= bi[q]);
    if (takeA) { o[t] = a[p]; oi[t] = ai[p]; ++p; }
    else       { o[t] = b[q]; oi[t] = bi[q]; ++q; }
  }
  for (int t = 0; t < TOPK; ++t) { a[t] = o[t]; ai[t] = oi[t]; }
}

// Stage 1: 32 blocks x 256 threads. Per-thread top-11 over a strided scan,
// then 8-level LDS tree merge of sorted lists -> one top-11 per block.
__global__ __launch_bounds__(S1_THREADS) void topk_stage1(
    const float* __restrict__ sim, float* __restrict__ cVals,
    int* __restrict__ cIds) {
  __shared__ float sV[S1_THREADS * TOPK];
  __shared__ int   sI[S1_THREADS * TOPK];

  float v[TOPK]; int id[TOPK];
  #pragma unroll
  for (int t = 0; t < TOPK; ++t) { v[t] = -3.4e38f; id[t] = 0x7fffffff; }

  const int tid = threadIdx.x;
  const int stride = S1_BLOCKS * S1_THREADS;
  for (int i = blockIdx.x * S1_THREADS + tid; i < VOCAB; i += stride)
    insert_desc(v, id, sim[i], i);

  for (int active = S1_THREADS >> 1; active >= 1; active >>= 1) {
    if (tid >= active && tid < 2 * active) {
      #pragma unroll
      for (int t = 0; t < TOPK; ++t) {
        sV[tid * TOPK + t] = v[t];
        sI[tid * TOPK + t] = id[t];
      }
    }
    __syncthreads();
    if (tid < active) {
      float b[TOPK]; int bi[TOPK];
      #pragma unroll
      for (int t = 0; t < TOPK; ++t) {
        b[t]  = sV[(tid + active) * TOPK + t];
        bi[t] = sI[(tid + active) * TOPK + t];
      }
      merge_desc(v, id, b, bi);
    }
    __syncthreads();
  }

  if (tid == 0) {
    for (int t = 0; t < TOPK; ++t) {
      cVals[blockIdx.x * TOPK + t] = v[t];
      cIds [blockIdx.x * TOPK + t] = id[t];
    }
  }
}

// Stage 2: one wave merges the 32 block candidates; thread 0 writes
// out[0..10] = top values, out[11..21] = top indices (as float).
__global__ __launch_bounds__(32) void topk_stage2(
    const float* __restrict__ cVals, const int* __restrict__ cIds,
    float* __restrict__ out) {
  __shared__ float sV[32 * TOPK];
  __shared__ int   sI[32 * TOPK];

  const int tid = threadIdx.x;
  float v[TOPK]; int id[TOPK];
  #pragma unroll
  for (int t = 0; t < TOPK; ++t) {
    v[t]  = cVals[tid * TOPK + t];
    id[t] = cIds [tid * TOPK + t];
  }

  for (int active = 16; active >= 1; active >>= 1) {
    if (tid >= active && tid < 2 * active) {
      #pragma unroll
      for (int t = 0; t < TOPK; ++t) {
        sV[tid * TOPK + t] = v[t];
        sI[tid * TOPK + t] = id[t];
      }
    }
    __syncthreads();
    if (tid < active) {
      float b[TOPK]; int bi[TOPK];
      #pragma unroll
      for (int t = 0; t < TOPK; ++t) {
        b[t]  = sV[(tid + active) * TOPK + t];
        bi[t] = sI[(tid + active) * TOPK + t];
      }
      merge_desc(v, id, b, bi);
    }
    __syncthreads();
  }

  if (tid == 0) {
    for (int t = 0; t < TOPK; ++t) {
      out[t]        = v[t];
      out[TOPK + t] = (float)id[t];
    }
  }
}

extern "C" void kernel_launch(void* const* d_in, const int* in_sizes, int n_in,
                              void* d_out, int out_size, void* d_ws, size_t ws_size,
                              hipStream_t stream) {
  (void)in_sizes; (void)n_in; (void)out_size; (void)ws_size;

  const int*   wid = (const int*)d_in[0];     // wordid (int32, 1 element)
  const float* emb = (const float*)d_in[1];   // embedding [200000 x 300] f32
  // d_in[2] = topk (==10); TOPK = topk+1 hardcoded to match reference shapes.

  // Workspace: 800000 B scores + 32*11 candidate (val,id) pairs.
  float* sim   = (float*)d_ws;
  float* cVals = (float*)((char*)d_ws + (size_t)VOCAB * sizeof(float));
  int*   cIds  = (int*)  ((char*)d_ws + (size_t)VOCAB * sizeof(float)
                                      + (size_t)S1_BLOCKS * TOPK * sizeof(float));

  // 16 rows/wave, 4 waves/block -> 64 rows/block; 200000/64 = 3125 blocks exact.
  gemv_wmma  <<<VOCAB / 64, 128, 0, stream>>>(emb, wid, sim);
  topk_stage1<<<S1_BLOCKS, S1_THREADS, 0, stream>>>(sim, cVals, cIds);
  topk_stage2<<<1, 32, 0, stream>>>(cVals, cIds, (float*)d_out);
}